// FlyingCartpole_44710609551637
// MI455X (gfx1250) — compile-verified
//
#include <hip/hip_runtime.h>

typedef float v2f __attribute__((ext_vector_type(2)));

namespace {

constexpr float kMass     = 2.1f;              // MASS_Q + MASS_P
constexpr float kInvMass  = 1.0f / 2.1f;
constexpr float kJx       = 0.0023f;
constexpr float kJz       = 0.004f;
constexpr float kJixInv   = 1.0f / 0.0023f;    // JINV diag xy
constexpr float kJizInv   = 250.0f;            // JINV diag z
constexpr float kGz       = -9.81f;
constexpr float kMotorD   = 0.175f;
constexpr float kKm       = 0.025f;
constexpr float kDt       = 0.05f;
constexpr float kDt2      = 0.025f;
constexpr float kDt6      = 0.05f / 6.0f;
constexpr float kActScale = 10.0f;
constexpr float kUHover   = 0.515025f;         // MASS*9.81/(10*1*4)
constexpr float kInvSqrt2 = 0.70710678118654752440f;
constexpr float kInvL     = 2.0f;              // 1/L

__device__ __forceinline__ v2f vsplat(float c) { v2f r; r.x = c; r.y = c; return r; }

// ---- CDNA5 packed fp32 (VOP3P) via inline asm; ISel won't form these. ----
__device__ __forceinline__ v2f pk_mul(v2f a, v2f b) {        // a*b
  v2f d; asm("v_pk_mul_f32 %0, %1, %2" : "=v"(d) : "v"(a), "v"(b)); return d;
}
__device__ __forceinline__ v2f pk_add(v2f a, v2f b) {        // a+b
  v2f d; asm("v_pk_add_f32 %0, %1, %2" : "=v"(d) : "v"(a), "v"(b)); return d;
}
__device__ __forceinline__ v2f pk_sub(v2f a, v2f b) {        // a-b
  v2f d; asm("v_pk_add_f32 %0, %1, %2 neg_lo:[0,1] neg_hi:[0,1]"
             : "=v"(d) : "v"(a), "v"(b)); return d;
}
__device__ __forceinline__ v2f pk_fma(v2f a, v2f b, v2f c) { // a*b+c
  v2f d; asm("v_pk_fma_f32 %0, %1, %2, %3" : "=v"(d) : "v"(a), "v"(b), "v"(c)); return d;
}
__device__ __forceinline__ v2f pk_fms(v2f a, v2f b, v2f c) { // a*b-c
  v2f d; asm("v_pk_fma_f32 %0, %1, %2, %3 neg_lo:[0,0,1] neg_hi:[0,0,1]"
             : "=v"(d) : "v"(a), "v"(b), "v"(c)); return d;
}
__device__ __forceinline__ v2f pk_fnma(v2f a, v2f b, v2f c) { // c-a*b
  v2f d; asm("v_pk_fma_f32 %0, %1, %2, %3 neg_lo:[1,0,0] neg_hi:[1,0,0]"
             : "=v"(d) : "v"(a), "v"(b), "v"(c)); return d;
}

// Per-component TRANS ops (no packed form for v_rcp/v_sin/v_cos).
__device__ __forceinline__ v2f rcp2(v2f a) {
  v2f r;
  r.x = __builtin_amdgcn_rcpf(a.x);
  r.y = __builtin_amdgcn_rcpf(a.y);
  return r;
}
__device__ __forceinline__ v2f sin2(v2f a) {
  v2f r; r.x = __sinf(a.x); r.y = __sinf(a.y); return r;
}
__device__ __forceinline__ v2f cos2(v2f a) {
  v2f r; r.x = __cosf(a.x); r.y = __cosf(a.y); return r;
}

// One evaluation of dynamics_(x, u) for two independent rows (packed lanes).
// State layout: p[0:3] m[3:6] theta[6] v[7:10] w[10:13] theta_dot[13]
__device__ __forceinline__ void dyn(const v2f s[14], v2f Fz,
                                    v2f tx, v2f ty, v2f tz,
                                    v2f d[14]) {
  const v2f m0 = s[3],  m1 = s[4],  m2 = s[5];
  const v2f th = s[6];
  const v2f v0 = s[7],  v1 = s[8],  v2 = s[9];
  const v2f w0 = s[10], w1 = s[11], w2 = s[12];

  const v2f one = vsplat(1.0f);
  const v2f two = vsplat(2.0f);

  // mrp2quat: n2 shared by q(m) and q(-m)
  const v2f n2   = pk_fma(m2, m2, pk_fma(m1, m1, pk_mul(m0, m0)));
  const v2f inv  = rcp2(pk_add(one, n2));       // v_rcp_f32 x2
  const v2f on   = pk_sub(one, n2);
  const v2f qs   = pk_mul(on, inv);
  const v2f inv2 = pk_add(inv, inv);
  const v2f q0 = pk_mul(m0, inv2);
  const v2f q1 = pk_mul(m1, inv2);
  const v2f q2 = pk_mul(m2, inv2);

  // f = (0,0,Fz) + quatrot(q(-m), (0,0,MASS*g)); signs of -q folded in.
  // t = cross(-q,(0,0,g)) + qs*(0,0,g) = (-q1*g, q0*g, qs*g)
  const v2f vg  = vsplat(kMass * kGz);
  const v2f vgn = vsplat(-kMass * kGz);
  const v2f t0 = pk_mul(q1, vgn);
  const v2f t1 = pk_mul(q0, vg);
  const v2f t2 = pk_mul(qs, vg);
  v2f f0 = pk_mul(two, pk_fnma(q1, t2, pk_mul(q2, t1)));    // 2*(q2*t1 - q1*t2)
  v2f f1 = pk_mul(two, pk_fnma(q2, t0, pk_mul(q0, t2)));    // 2*(q0*t2 - q2*t0)
  v2f f2 = pk_fma(two, pk_fnma(q0, t1, pk_mul(q1, t0)), vg); // g + 2*(q1*t0 - q0*t1)
  f2 = pk_add(f2, Fz);

  // mdot = 0.25*((1-n2)*w + 2*cross(m,w) + 2*m*(m.w))
  const v2f quarter = vsplat(0.25f);
  const v2f mw  = pk_fma(m2, w2, pk_fma(m1, w1, pk_mul(m0, w0)));
  const v2f mw2 = pk_add(mw, mw);
  const v2f cx = pk_fnma(m2, w1, pk_mul(m1, w2));
  const v2f cy = pk_fnma(m0, w2, pk_mul(m2, w0));
  const v2f cz = pk_fnma(m1, w0, pk_mul(m0, w1));
  d[3] = pk_mul(quarter, pk_fma(m0, mw2, pk_fma(two, cx, pk_mul(on, w0))));
  d[4] = pk_mul(quarter, pk_fma(m1, mw2, pk_fma(two, cy, pk_mul(on, w1))));
  d[5] = pk_mul(quarter, pk_fma(m2, mw2, pk_fma(two, cz, pk_mul(on, w2))));

  // pdot = quatrot(q, v) = v + 2*cross(qv, cross(qv,v)+qs*v)
  {
    const v2f p0 = pk_fma(qs, v0, pk_fnma(q2, v1, pk_mul(q1, v2)));
    const v2f p1 = pk_fma(qs, v1, pk_fnma(q0, v2, pk_mul(q2, v0)));
    const v2f p2 = pk_fma(qs, v2, pk_fnma(q1, v0, pk_mul(q0, v1)));
    d[0] = pk_fma(two, pk_fnma(q2, p1, pk_mul(q1, p2)), v0);
    d[1] = pk_fma(two, pk_fnma(q0, p2, pk_mul(q2, p0)), v1);
    d[2] = pk_fma(two, pk_fnma(q1, p0, pk_mul(q0, p1)), v2);
  }

  // vdot = f/MASS - cross(w, v)
  const v2f invm = vsplat(kInvMass);
  const v2f wxv0 = pk_fnma(w2, v1, pk_mul(w1, v2));
  const v2f wxv1 = pk_fnma(w0, v2, pk_mul(w2, v0));
  const v2f wxv2 = pk_fnma(w1, v0, pk_mul(w0, v1));
  const v2f vd0 = pk_fms(f0, invm, wxv0);
  const v2f vd1 = pk_fms(f1, invm, wxv1);
  const v2f vd2 = pk_fms(f2, invm, wxv2);
  d[7] = vd0; d[8] = vd1; d[9] = vd2;

  // wdot = JINV*(tau - cross(w, J*w)); J, JINV diagonal
  const v2f jx = vsplat(kJx), jz = vsplat(kJz);
  const v2f jix = vsplat(kJixInv), jiz = vsplat(kJizInv);
  const v2f Jw0 = pk_mul(jx, w0), Jw1 = pk_mul(jx, w1), Jw2 = pk_mul(jz, w2);
  const v2f cj0 = pk_fnma(w2, Jw1, pk_mul(w1, Jw2));
  const v2f cj1 = pk_fnma(w0, Jw2, pk_mul(w2, Jw0));
  const v2f cj2 = pk_fnma(w1, Jw0, pk_mul(w0, Jw1));
  d[10] = pk_mul(jix, pk_sub(tx, cj0));
  d[11] = pk_mul(jix, pk_sub(ty, cj1));
  d[12] = pk_mul(jiz, pk_sub(tz, cj2));

  // theta dynamics; only component 0 of quatrot(q, vdot) needed.
  d[6] = s[13];
  const v2f a0 = pk_fma(qs, vd0, pk_fnma(q2, vd1, pk_mul(q1, vd2)));
  const v2f a1 = pk_fma(qs, vd1, pk_fnma(q0, vd2, pk_mul(q2, vd0)));
  const v2f a2 = pk_fma(qs, vd2, pk_fnma(q1, vd0, pk_mul(q0, vd1)));
  const v2f xdd = pk_fma(two, pk_fnma(q2, a1, pk_mul(q1, a2)), vd0);
  const v2f gzv = vsplat(kGz), invLv = vsplat(kInvL);
  d[13] = pk_mul(invLv, pk_fma(gzv, sin2(th), pk_mul(xdd, cos2(th))));
}

// Full RK4 step for two rows in lockstep. u-derived terms are RK4-invariant.
__device__ __forceinline__ void rk4_pair(const v2f s[14],
                                         v2f u0, v2f u1, v2f u2, v2f u3,
                                         v2f out[14]) {
  const v2f scl = vsplat(kActScale);
  const v2f hvs = vsplat(kActScale * kUHover);
  const v2f up0 = pk_fma(scl, u0, hvs);
  const v2f up1 = pk_fma(scl, u1, hvs);
  const v2f up2 = pk_fma(scl, u2, hvs);
  const v2f up3 = pk_fma(scl, u3, hvs);
  const v2f Fz  = pk_add(pk_add(up0, up1), pk_add(up2, up3));   // KF = 1
  const v2f ddv = vsplat(kMotorD * kInvSqrt2);
  const v2f kmv = vsplat(kKm);
  const v2f tx = pk_mul(ddv, pk_add(pk_sub(up0, up1), pk_sub(up3, up2)));
  const v2f ty = pk_mul(ddv, pk_add(pk_sub(up2, up0), pk_sub(up3, up1)));
  const v2f tz = pk_mul(kmv, pk_sub(pk_add(up0, up2), pk_add(up1, up3)));

  const v2f two  = vsplat(2.0f);
  const v2f dt2v = vsplat(kDt2);
  const v2f dtv  = vsplat(kDt);
  const v2f dt6v = vsplat(kDt6);

  v2f k[14], acc[14], tmp[14];
  dyn(s, Fz, tx, ty, tz, k);                            // k1
#pragma unroll
  for (int i = 0; i < 14; ++i) { acc[i] = k[i]; tmp[i] = pk_fma(dt2v, k[i], s[i]); }
  dyn(tmp, Fz, tx, ty, tz, k);                          // k2
#pragma unroll
  for (int i = 0; i < 14; ++i) { acc[i] = pk_fma(two, k[i], acc[i]); tmp[i] = pk_fma(dt2v, k[i], s[i]); }
  dyn(tmp, Fz, tx, ty, tz, k);                          // k3
#pragma unroll
  for (int i = 0; i < 14; ++i) { acc[i] = pk_fma(two, k[i], acc[i]); tmp[i] = pk_fma(dtv, k[i], s[i]); }
  dyn(tmp, Fz, tx, ty, tz, k);                          // k4
#pragma unroll
  for (int i = 0; i < 14; ++i) { out[i] = pk_fma(dt6v, pk_add(acc[i], k[i]), s[i]); }
}

// Two rows per lane: 2 rows of x = 112 B = 7 aligned float4 -> all global
// traffic is b128; the two rows ride in the lo/hi halves of a VGPR pair so
// the math issues as v_pk_*_f32 (2 fp32 FLOPs per VALU issue).
__global__ void __launch_bounds__(256) fc_rk4_pairs(
    const float4* __restrict__ x4, const float4* __restrict__ u4,
    float4* __restrict__ o4, unsigned npairs) {
  const unsigned t = blockIdx.x * blockDim.x + threadIdx.x;
  if (t >= npairs) return;

  float4 a[7];
#pragma unroll
  for (unsigned i = 0; i < 7; ++i) a[i] = x4[t * 7u + i];  // global_load_b128
  const float4 ua = u4[t * 2u + 0u];
  const float4 ub = u4[t * 2u + 1u];

  // Transpose the 2x14 tile into 14 x v2f (register shuffles only).
  const float* ap = reinterpret_cast<const float*>(a);
  v2f s[14];
#pragma unroll
  for (int j = 0; j < 14; ++j) { s[j].x = ap[j]; s[j].y = ap[14 + j]; }

  v2f u0; u0.x = ua.x; u0.y = ub.x;
  v2f u1; u1.x = ua.y; u1.y = ub.y;
  v2f u2; u2.x = ua.z; u2.y = ub.z;
  v2f u3; u3.x = ua.w; u3.y = ub.w;

  v2f r[14];
  rk4_pair(s, u0, u1, u2, u3, r);

  float rr[28];
#pragma unroll
  for (int j = 0; j < 14; ++j) { rr[j] = r[j].x; rr[14 + j] = r[j].y; }
#pragma unroll
  for (unsigned i = 0; i < 7; ++i) {
    float4 o;
    o.x = rr[4 * i + 0]; o.y = rr[4 * i + 1];
    o.z = rr[4 * i + 2]; o.w = rr[4 * i + 3];
    o4[t * 7u + i] = o;                                    // global_store_b128
  }
}

// Tail kernel for odd B (not hit for B = 2^21, kept for generality).
__global__ void fc_rk4_tail(const float* __restrict__ x, const float* __restrict__ u,
                            float* __restrict__ o, int row) {
  if (threadIdx.x != 0 || blockIdx.x != 0) return;
  v2f s[14], r[14];
#pragma unroll
  for (int i = 0; i < 14; ++i) {
    const float v = x[row * 14 + i];
    s[i].x = v; s[i].y = v;
  }
  v2f u0 = vsplat(u[row * 4 + 0]);
  v2f u1 = vsplat(u[row * 4 + 1]);
  v2f u2 = vsplat(u[row * 4 + 2]);
  v2f u3 = vsplat(u[row * 4 + 3]);
  rk4_pair(s, u0, u1, u2, u3, r);
#pragma unroll
  for (int i = 0; i < 14; ++i) o[row * 14 + i] = r[i].x;
}

}  // namespace

extern "C" void kernel_launch(void* const* d_in, const int* in_sizes, int n_in,
                              void* d_out, int out_size, void* d_ws, size_t ws_size,
                              hipStream_t stream) {
  const float* x = (const float*)d_in[0];   // (B, 14) f32
  const float* u = (const float*)d_in[1];   // (B, 4)  f32
  float* out = (float*)d_out;               // (B, 14) f32

  const long long B = (long long)in_sizes[0] / 14;
  const unsigned npairs = (unsigned)(B / 2);
  if (npairs > 0) {
    const unsigned block = 256;             // 8 wave32 per block
    const unsigned grid  = (npairs + block - 1) / block;
    fc_rk4_pairs<<<dim3(grid), dim3(block), 0, stream>>>(
        (const float4*)x, (const float4*)u, (float4*)out, npairs);
  }
  if (B & 1) {
    fc_rk4_tail<<<1, 1, 0, stream>>>(x, u, out, (int)(B - 1));
  }
}